// PatchMatchingLoRA_63350767616602
// MI455X (gfx1250) — compile-verified
//
#include <hip/hip_runtime.h>
#include <math.h>

#define NUM_CLASSES 1000
#define KF 16          // NUM_FINE
#define DIM 768        // TEXT_DIM
#define BATCH 128
#define NPATCH 576
#define INV_EPS 20.0f  // 1/0.05
#define NITERS 3

typedef float v2f __attribute__((ext_vector_type(2)));
typedef float v8f __attribute__((ext_vector_type(8)));

#define TC_PITCH 772   // 772 mod 64 == 4 -> conflict-free B-fragment reads
#define S_PITCH  20    // 8*20 mod 64 == 32 -> half-waves hit disjoint banks

__launch_bounds__(256, 2)
__global__ void patch_match_kernel(const float* __restrict__ img,
                                   const float* __restrict__ text,
                                   const float* __restrict__ fw,
                                   const int* __restrict__ tgt,
                                   float* __restrict__ out_label,
                                   float* __restrict__ out_target)
{
    __shared__ float tc[KF][TC_PITCH];       // text_c[k][d]
    __shared__ float S[NPATCH][S_PITCH];     // exp(scores/eps), [patch][k]
    __shared__ float psum[256];
    __shared__ float rsc[KF];                // row (per-k) scale
    __shared__ float csc[NPATCH];            // col (per-patch) scale

    const int b   = blockIdx.x;
    const int tid = threadIdx.x;
    const int cls = tgt[b];

    // ---- Phase 1: text_c = text[cls] + fine_weights[cls] -> LDS ----
    for (int idx = tid; idx < KF * DIM; idx += 256) {
        const int k = idx / DIM;
        const int d = idx - k * DIM;
        tc[k][d] = text[(size_t)cls * DIM + d] + fw[((size_t)cls * KF + k) * DIM + d];
    }
    __syncthreads();

    // ---- Phase 2: scores[576][16] via V_WMMA_F32_16X16X4_F32 ----
    const int lane = tid & 31;
    const int wave = tid >> 5;     // wave-uniform
    const int half = lane >> 4;    // 0: K=0,1   1: K=2,3
    const int l16  = lane & 15;

    for (int tile = wave; tile < NPATCH / 16; tile += 8) {
        const int row = tile * 16 + l16;               // A-matrix row (M = l16)
        const float* aptr = img + ((size_t)b * NPATCH + row) * DIM + half * 2;
        const float* bptr = &tc[l16][half * 2];        // B col n = l16
        v8f acc = {0.f, 0.f, 0.f, 0.f, 0.f, 0.f, 0.f, 0.f};
        #pragma unroll 8
        for (int kd = 0; kd < DIM; kd += 4) {
            v2f a  = *(const v2f*)(aptr + kd);         // A[M=row][kd+2*half .. +1]
            v2f bb = *(const v2f*)(bptr + kd);         // B[kd+2*half..][N=l16]
            acc = __builtin_amdgcn_wmma_f32_16x16x4_f32(
                      false, a, false, bb, (short)0, acc, false, false);
        }
        // C/D layout: acc[i] = score[M = i + 8*half][N = l16]
        #pragma unroll
        for (int i = 0; i < 8; ++i) {
            const int r = tile * 16 + i + half * 8;
            S[r][l16] = expf(acc[i] * INV_EPS);
        }
    }
    __syncthreads();

    // ---- Phase 3: Sinkhorn with multiplicative factors ----
    // Q[k][n] = S[n][k] * r[k] * c[n]; global normalization cancels in first row step.
    for (int n = tid; n < NPATCH; n += 256) csc[n] = 1.0f;
    __syncthreads();

    const int kk = tid & 15;
    const int g  = tid >> 4;   // 16 row-groups
    for (int it = 0; it < NITERS; ++it) {
        // row step: r[k] = 1 / (K * sum_n S[n][k]*c[n])
        float p = 0.f;
        for (int n = g; n < NPATCH; n += 16) p += S[n][kk] * csc[n];
        psum[tid] = p;
        __syncthreads();
        if (tid < KF) {
            float rs = 0.f;
            #pragma unroll
            for (int j = 0; j < 16; ++j) rs += psum[j * 16 + tid];
            rsc[tid] = 1.0f / ((float)KF * rs);
        }
        __syncthreads();
        // col step: c[n] = 1 / (N * sum_k S[n][k]*r[k])
        for (int n = tid; n < NPATCH; n += 256) {
            float s = 0.f;
            #pragma unroll
            for (int k2 = 0; k2 < KF; ++k2) s += S[n][k2] * rsc[k2];
            csc[n] = 1.0f / ((float)NPATCH * s);
        }
        __syncthreads();
    }

    // ---- Phase 4: fine_label = S*r*c*N, argmax -> fine_target ----
    for (int n = tid; n < NPATCH; n += 256) {
        const float cn = csc[n] * (float)NPATCH;
        float q[KF];
        float best = -1.0f;
        int bi = 0;
        #pragma unroll
        for (int k2 = 0; k2 < KF; ++k2) {
            const float v = S[n][k2] * rsc[k2] * cn;
            q[k2] = v;
            if (v > best) { best = v; bi = k2; }   // first-max tie-break, like argmax
        }
        float* dst = out_label + ((size_t)b * NPATCH + n) * KF;
        #pragma unroll
        for (int k2 = 0; k2 < KF; k2 += 4) {
            float4 vv = make_float4(q[k2], q[k2 + 1], q[k2 + 2], q[k2 + 3]);
            *(float4*)(dst + k2) = vv;
        }
        out_target[(size_t)b * NPATCH + n] = (float)(bi * NUM_CLASSES + cls);
    }
}

extern "C" void kernel_launch(void* const* d_in, const int* in_sizes, int n_in,
                              void* d_out, int out_size, void* d_ws, size_t ws_size,
                              hipStream_t stream) {
    (void)in_sizes; (void)n_in; (void)out_size; (void)d_ws; (void)ws_size;
    const float* img  = (const float*)d_in[0];   // [128,576,768] f32
    const float* text = (const float*)d_in[1];   // [1000,768] f32
    const float* fw   = (const float*)d_in[2];   // [1000,16,768] f32
    const int*   tgt  = (const int*)d_in[3];     // [128] int
    float* out_label  = (float*)d_out;                              // [128*576*16]
    float* out_tgt    = out_label + (size_t)BATCH * NPATCH * KF;    // [128*576]
    patch_match_kernel<<<BATCH, 256, 0, stream>>>(img, text, fw, tgt, out_label, out_tgt);
}